// degree_based_sampler_80994493268499
// MI455X (gfx1250) — compile-verified
//
#include <hip/hip_runtime.h>
#include <hip/hip_bf16.h>

// ---------------------------------------------------------------------------
// degree_based_sampler for MI455X (gfx1250)
//
// Pipeline (all on `stream`, capture-safe):
//   1. k_init      : zero probs / masks
//   2. k_centers   : per-class sampling w/o replacement via cycle-walking
//                    Feistel permutation; sets center/retained masks, writes
//                    center node ids (as float) to d_out.
//   3. k_edges     : ONE fused streaming pass over all edges, staged through
//                    LDS by the Tensor Data Mover (double-buffered
//                    tensor_load_to_lds + s_wait_tensorcnt). Does
//                    atomicAdd(probs[dst], w) and nbr[dst] |= center[src].
//   4. per hop (x2): k_reset -> k_keys (Gumbel keys for candidates)
//                    -> 4x (k_hist 256-bin LDS histogram + k_scan radix step)
//                    -> k_select (threshold + tie-break -> retained)
//   5. k_finalize  : retained (0/1) and masked probs to d_out.
// ---------------------------------------------------------------------------

#define NN        200000      // N_NODES (reference constant)
#define NCLS      10
#define TILE      2048        // edges per TDM tile
#define EDGE_GRID 640         // blocks for the edge pass
#define TPB       256

typedef unsigned int v4u __attribute__((ext_vector_type(4)));
typedef int          v8i __attribute__((ext_vector_type(8)));
typedef int          v4i __attribute__((ext_vector_type(4)));

struct SelState {
    unsigned prefix;      // threshold bits accumulated (radix select)
    unsigned k_rem;       // remaining k within current prefix bin
    unsigned select_all;  // 1 if #candidates <= k
    unsigned tie;         // tie-break counter
    unsigned cand_count;  // total candidates this hop
};

// ---- small deterministic mixers (counter-based PRNG) ----------------------
__device__ __forceinline__ unsigned mixu(unsigned h) {
    h ^= h >> 16; h *= 0x85EBCA6Bu;
    h ^= h >> 13; h *= 0xC2B2AE35u;
    h ^= h >> 16; return h;
}
__device__ __forceinline__ unsigned rng32(unsigned seed, unsigned i) {
    return mixu(seed ^ mixu(i * 0x9E3779B9u + 0x7F4A7C15u));
}

// ---- Tensor Data Mover: 1-D tile load (global -> LDS), D# packed per ISA --
// amdgpu-toolchain (clang-23) builtin arity: 6 args
//   (uint32x4 g0, int32x8 g1, int32x4 g2, int32x4 g3, int32x8 gx, i32 cpol)
__device__ __forceinline__ void tdm_load_1d(unsigned lds_addr,
                                            const void* gptr,
                                            unsigned remaining_elems) {
    unsigned long long ga = (unsigned long long)(size_t)gptr;
    v4u g0;
    g0.x = 1u;                                         // count=1, user desc
    g0.y = lds_addr;                                   // LDS byte address
    g0.z = (unsigned)(ga & 0xFFFFFFFFull);             // global_addr[31:0]
    g0.w = (unsigned)((ga >> 32) & 0x1FFFFFFull)       // global_addr[56:32]
         | (2u << 30);                                 // type=2 ("image")
    unsigned d0 = remaining_elems;                     // tensor_dim0 (OOB->0)
    v8i g1;
    g1.s0 = (int)(2u << 16);                           // data_size=4B, mask=0
    g1.s1 = (int)((d0 & 0xFFFFu) << 16);               // tensor_dim0[15:0]
    g1.s2 = (int)(((d0 >> 16) & 0xFFFFu) | (1u << 16));// dim0[31:16], dim1=1
    g1.s3 = (int)((unsigned)TILE << 16);               // tile_dim0 = TILE
    g1.s4 = 1;                                         // tile_dim1=1 (one row)
    g1.s5 = (int)d0;                                   // dim0_stride[31:0]
    g1.s6 = 0;                                         // stride hi bits
    g1.s7 = 0;
    v4i z4 = {0, 0, 0, 0};
    v8i z8 = {0, 0, 0, 0, 0, 0, 0, 0};
    __builtin_amdgcn_tensor_load_to_lds(g0, g1, z4, z4, z8, 0);
}

// ---------------------------------------------------------------------------
__global__ void k_init(float* probs, unsigned char* center_mask,
                       unsigned char* nbr, unsigned char* retained, int n) {
    for (int i = blockIdx.x * blockDim.x + threadIdx.x; i < n;
         i += gridDim.x * blockDim.x) {
        probs[i] = 0.0f;
        center_mask[i] = 0;
        nbr[i] = 0;
        retained[i] = 0;
    }
}

// Sampling without replacement == first `budget` entries of a random
// permutation. Cycle-walking Feistel (16-bit, 4 rounds) gives a bijection
// on [0, P) per class, fully parallel and deterministic.
__global__ void k_centers(const int* ids, const int* budget_ptr,
                          unsigned char* center_mask, unsigned char* retained,
                          float* out_centers, int C, int P) {
    int j = blockIdx.x * blockDim.x + threadIdx.x;
    if (j >= C * P) return;
    int c = j / P, idx = j - c * P;
    int budget = *budget_ptr;
    if (idx >= budget) return;
    unsigned x = (unsigned)idx;
    do {
        unsigned L = (x >> 8) & 255u, R = x & 255u;
        #pragma unroll
        for (int r = 0; r < 4; ++r) {
            unsigned f = mixu(R ^ (unsigned)(c * 0x9E37) ^ (unsigned)(r * 0x85EB)
                              ^ 0xA5A5A5A5u) & 255u;
            unsigned t = L ^ f;
            L = R; R = t;
        }
        x = (L << 8) | R;
    } while (x >= (unsigned)P);
    int node = ids[c * P + (int)x];
    center_mask[node] = 1;
    retained[node]    = 1;
    out_centers[c * budget + idx] = (float)node;
}

// Fused degree + neighbor-mask pass over all edges, TDM double-buffered.
__global__ void __launch_bounds__(TPB)
k_edges(const int* __restrict__ src, const int* __restrict__ dst,
        const float* __restrict__ w, const unsigned char* __restrict__ center_mask,
        float* probs, unsigned char* nbr, int E, int tiles_per_block, int n_tiles) {
    __shared__ int   s_src[2][TILE];
    __shared__ int   s_dst[2][TILE];
    __shared__ float s_w  [2][TILE];

    int t0 = blockIdx.x * tiles_per_block;
    int t1 = t0 + tiles_per_block;
    if (t1 > n_tiles) t1 = n_tiles;
    if (t0 >= t1) return;

    if (threadIdx.x == 0) {                       // preload tile t0 -> buf 0
        int base = t0 * TILE;
        unsigned rem = (unsigned)(E - base);
        tdm_load_1d((unsigned)(size_t)&s_src[0][0], src + base, rem);
        tdm_load_1d((unsigned)(size_t)&s_dst[0][0], dst + base, rem);
        tdm_load_1d((unsigned)(size_t)&s_w[0][0],   w   + base, rem);
    }
    for (int t = t0; t < t1; ++t) {
        int buf = (t - t0) & 1;
        if (threadIdx.x == 0) {
            if (t + 1 < t1) {                     // prefetch next tile
                int base = (t + 1) * TILE;
                unsigned rem = (unsigned)(E - base);
                int nb = buf ^ 1;
                tdm_load_1d((unsigned)(size_t)&s_src[nb][0], src + base, rem);
                tdm_load_1d((unsigned)(size_t)&s_dst[nb][0], dst + base, rem);
                tdm_load_1d((unsigned)(size_t)&s_w[nb][0],   w   + base, rem);
                __builtin_amdgcn_s_wait_tensorcnt(3);  // tile t's 3 loads done
            } else {
                __builtin_amdgcn_s_wait_tensorcnt(0);
            }
        }
        __syncthreads();
        int base = t * TILE;
        for (int i = threadIdx.x; i < TILE; i += blockDim.x) {
            int e = base + i;
            if (e < E) {
                int   d  = s_dst[buf][i];
                float wt = s_w[buf][i];
                atomicAdd(&probs[d], wt);          // in-degree (L2-resident)
                int s = s_src[buf][i];
                if (center_mask[s]) nbr[d] = 1;    // benign write race
            }
        }
        __syncthreads();
    }
}

__global__ void k_reset(SelState* st, unsigned* hist) {
    if (blockIdx.x == 0) {
        if (threadIdx.x < 256) hist[threadIdx.x] = 0;
        if (threadIdx.x == 0) {
            st->prefix = 0; st->k_rem = 0; st->select_all = 0;
            st->tie = 0;    st->cand_count = 0;
        }
    }
}

// Gumbel keys: key = log(probs) + Gumbel; mapped to order-preserving u32.
// Every candidate has in-degree >= 1 so keys are always finite.
__global__ void k_keys(const float* probs, const unsigned char* nbr,
                       const unsigned char* retained, unsigned char* cand,
                       unsigned* keys, SelState* st, unsigned hop_seed, int n) {
    for (int i = blockIdx.x * blockDim.x + threadIdx.x; i < n;
         i += gridDim.x * blockDim.x) {
        bool c = nbr[i] && !retained[i];
        cand[i] = c ? 1 : 0;
        if (c) {
            unsigned r = rng32(hop_seed, (unsigned)i);
            float U = (float)(r >> 8) * (1.0f / 16777216.0f)
                    + (0.5f / 16777216.0f);              // U in (0,1)
            float g   = -__logf(-__logf(U));             // Gumbel(0,1)
            float key = __logf(probs[i]) + g;
            unsigned u = __float_as_uint(key);
            u = (u & 0x80000000u) ? ~u : (u | 0x80000000u);
            keys[i] = u;
            atomicAdd(&st->cand_count, 1u);
        }
    }
}

// One 8-bit radix-select level: 256-bin histogram (LDS ds_add, then merge).
__global__ void k_hist(const unsigned char* cand, const unsigned* keys,
                       unsigned* hist, const SelState* st, int level, int n) {
    __shared__ unsigned lh[256];
    if (threadIdx.x < 256) lh[threadIdx.x] = 0;
    __syncthreads();
    int shift = 24 - 8 * level;
    unsigned pfx = st->prefix;
    for (int i = blockIdx.x * blockDim.x + threadIdx.x; i < n;
         i += gridDim.x * blockDim.x) {
        if (cand[i]) {
            unsigned u = keys[i];
            if (level == 0 || ((u >> (shift + 8)) == (pfx >> (shift + 8)))) {
                atomicAdd(&lh[(u >> shift) & 255u], 1u);
            }
        }
    }
    __syncthreads();
    if (threadIdx.x < 256 && lh[threadIdx.x])
        atomicAdd(&hist[threadIdx.x], lh[threadIdx.x]);
}

// Serial 256-bin scan (descending) -> extend prefix, update k_rem, zero hist.
__global__ void k_scan(SelState* st, unsigned* hist, const int* k_ptr, int level) {
    if (blockIdx.x != 0 || threadIdx.x != 0) return;
    if (level == 0) {
        unsigned k = (unsigned)*k_ptr;
        if (st->cand_count <= k) st->select_all = 1;
        st->k_rem = k;
    }
    if (!st->select_all) {
        unsigned k = st->k_rem, cum = 0;
        int b;
        for (b = 255; b >= 0; --b) {
            unsigned h = hist[b];
            if (cum + h >= k) break;
            cum += h;
        }
        if (b < 0) b = 0;
        st->prefix |= ((unsigned)b) << (24 - 8 * level);
        st->k_rem = k - cum;
    }
    for (int i = 0; i < 256; ++i) hist[i] = 0;
}

// Accept u > T, plus first k_rem ties at u == T (atomic tie-break).
__global__ void k_select(const unsigned char* cand, const unsigned* keys,
                         unsigned char* retained, SelState* st, int n) {
    for (int i = blockIdx.x * blockDim.x + threadIdx.x; i < n;
         i += gridDim.x * blockDim.x) {
        if (!cand[i]) continue;
        if (st->select_all) { retained[i] = 1; continue; }
        unsigned u = keys[i], T = st->prefix;
        if (u > T) {
            retained[i] = 1;
        } else if (u == T) {
            unsigned p = atomicAdd(&st->tie, 1u);
            if (p < st->k_rem) retained[i] = 1;
        }
    }
}

__global__ void k_finalize(const unsigned char* retained, const float* probs,
                           float* out, int centers_len, int n) {
    for (int i = blockIdx.x * blockDim.x + threadIdx.x; i < n;
         i += gridDim.x * blockDim.x) {
        bool r = retained[i] != 0;
        out[centers_len + i]     = r ? 1.0f : 0.0f;
        out[centers_len + n + i] = r ? probs[i] : 0.0f;
    }
}

// ---------------------------------------------------------------------------
static inline size_t align256(size_t x) { return (x + 255) & ~(size_t)255; }

extern "C" void kernel_launch(void* const* d_in, const int* in_sizes, int n_in,
                              void* d_out, int out_size, void* d_ws, size_t ws_size,
                              hipStream_t stream) {
    const int*   edge_index = (const int*)d_in[0];   // [2, E]
    const float* weights    = (const float*)d_in[1]; // [E]
    const int*   ids        = (const int*)d_in[2];   // [C, P]
    const int*   budget_ptr = (const int*)d_in[3];   // scalar
    const int*   hop1_ptr   = (const int*)d_in[4];   // scalar
    const int*   hop2_ptr   = (const int*)d_in[5];   // scalar

    const int E = in_sizes[0] / 2;
    const int C = NCLS;
    const int P = in_sizes[2] / C;
    const int N = NN;
    const int centers_len = out_size - 2 * N;        // == C * budget

    const int* src = edge_index;
    const int* dst = edge_index + E;
    float* out = (float*)d_out;

    // workspace carve-up
    char* ws = (char*)d_ws;
    float* probs = (float*)ws;                 ws += align256((size_t)N * 4);
    unsigned char* center_mask = (unsigned char*)ws; ws += align256((size_t)N);
    unsigned char* nbr      = (unsigned char*)ws;    ws += align256((size_t)N);
    unsigned char* retained = (unsigned char*)ws;    ws += align256((size_t)N);
    unsigned char* cand     = (unsigned char*)ws;    ws += align256((size_t)N);
    unsigned* keys = (unsigned*)ws;            ws += align256((size_t)N * 4);
    unsigned* hist = (unsigned*)ws;            ws += align256(256 * 4);
    SelState* st   = (SelState*)ws;            ws += align256(sizeof(SelState));
    (void)ws_size; (void)n_in;

    const int gN = (N + TPB - 1) / TPB;

    k_init<<<gN, TPB, 0, stream>>>(probs, center_mask, nbr, retained, N);

    const int gCP = (C * P + TPB - 1) / TPB;
    k_centers<<<gCP, TPB, 0, stream>>>(ids, budget_ptr, center_mask, retained,
                                       out, C, P);

    const int n_tiles = (E + TILE - 1) / TILE;
    const int tpb_tiles = (n_tiles + EDGE_GRID - 1) / EDGE_GRID;
    k_edges<<<EDGE_GRID, TPB, 0, stream>>>(src, dst, weights, center_mask,
                                           probs, nbr, E, tpb_tiles, n_tiles);

    const unsigned hop_seeds[2] = {0xC0FFEE42u, 0xBADC0DE7u};
    const int* hop_k[2] = {hop1_ptr, hop2_ptr};
    for (int hop = 0; hop < 2; ++hop) {
        k_reset<<<1, 256, 0, stream>>>(st, hist);
        k_keys<<<gN, TPB, 0, stream>>>(probs, nbr, retained, cand, keys, st,
                                       hop_seeds[hop], N);
        for (int level = 0; level < 4; ++level) {
            k_hist<<<gN, TPB, 0, stream>>>(cand, keys, hist, st, level, N);
            k_scan<<<1, 32, 0, stream>>>(st, hist, hop_k[hop], level);
        }
        k_select<<<gN, TPB, 0, stream>>>(cand, keys, retained, st, N);
    }

    k_finalize<<<gN, TPB, 0, stream>>>(retained, probs, out, centers_len, N);
}